// _CubeEngineCell_89060441850384
// MI455X (gfx1250) — compile-verified
//
#include <hip/hip_runtime.h>
#include <math.h>

// ---------------------------------------------------------------------------
// CubeEngineCell on MI455X (gfx1250, wave32)
//   - one workgroup per (b,r) cube (256 blocks x 1024 threads = 32 waves)
//   - all 18 mode contractions done with V_WMMA_F32_16X16X4_F32 on an
//     LDS-resident padded cube (in-place, register accumulators)
//   - MODE is a template parameter -> branch-free affine LDS addressing
//   - complex chain uses two LDS cubes (zr, zi): 2 x 144.5 KB < 320 KB WGP LDS
//   - zr and (coupled+zr) staged through d_out's new_h region (same-thread RMW)
//   - fused elementwise + per-cube RMS reduction + acc update at the end
// ---------------------------------------------------------------------------

typedef float v2f __attribute__((ext_vector_type(2)));
typedef float v8f __attribute__((ext_vector_type(8)));

#define NT        1024
#define CUBE      32768
#define LS1       36          // padded middle-axis stride (floats): 4*odd mod 64 banks
#define LS0       1156        // padded outer-axis stride  (floats): 4*odd mod 64 banks
#define CUBE_PAD  36984       // 31*LS0 + 31*LS1 + 32 floats = 147,936 B (16B aligned)
#define ETA       0.1f
#define MAGD      0.9f

__device__ __forceinline__ int ldsLin(int i) {
  return (i >> 10) * LS0 + ((i >> 5) & 31) * LS1 + (i & 31);
}

// p = flattened passive axes (0..1023), q = contracted-axis index (0..31)
template <int MODE>
__device__ __forceinline__ int ldsMode(int p, int q) {
  const int ph = p >> 5, pl = p & 31;
  if (MODE == 0) return q  * LS0 + ph * LS1 + pl;
  if (MODE == 1) return ph * LS0 + q  * LS1 + pl;
  return              ph * LS0 + pl * LS1 + q;
}

template <int MODE> struct QS { static constexpr int v = (MODE == 0) ? LS0 : (MODE == 1) ? LS1 : 1; };
// stride along output row index v inside storeTile (p moves within a 32-block)
template <int MODE> struct PS { static constexpr int v = (MODE == 2) ? LS1 : 1; };

__device__ __forceinline__ v8f wmma4(v2f a, v2f b, v8f c) {
  // D = A(16x4,f32) * B(4x16,f32) + C(16x16,f32)
  return __builtin_amdgcn_wmma_f32_16x16x4_f32(false, a, false, b, (short)0, c,
                                               false, false);
}

// B fragment: VGPR0 holds K=k0 (lanes 0-15) / K=k0+2 (lanes 16-31), VGPR1 = +1
__device__ __forceinline__ void loadB(const float* __restrict__ W, int ntile,
                                      int lane, v2f b[8]) {
  const int roff = (lane >= 16) ? 2 : 0;
  const int col  = ntile * 16 + (lane & 15);
#pragma unroll
  for (int k = 0; k < 8; ++k) {
    b[k].x = W[(k * 4 + roff)     * 32 + col];
    b[k].y = W[(k * 4 + roff + 1) * 32 + col];
  }
}

// K-loop of one 16x16 output tile: c += cube_tile(mtile) x B
template <int MODE>
__device__ __forceinline__ void tileAccum(const float* cube, int mtile, int lane,
                                          const v2f b[8], v8f& c) {
  const int p    = mtile * 16 + (lane & 15);
  const int qoff = (lane >= 16) ? 2 : 0;
  const int base = ldsMode<MODE>(p, qoff);
#pragma unroll
  for (int k = 0; k < 8; ++k) {
    v2f a;
    a.x = cube[base + (4 * k)     * QS<MODE>::v];
    a.y = cube[base + (4 * k + 1) * QS<MODE>::v];
    c = wmma4(a, b[k], c);
  }
}

// D fragment: VGPR v holds row M=v (lanes 0-15) / M=v+8 (lanes 16-31), N=lane&15
template <int MODE>
__device__ __forceinline__ void storeTile(float* cube, int mtile, int ntile,
                                          int lane, v8f d) {
  const int p0   = mtile * 16 + ((lane >= 16) ? 8 : 0);
  const int n    = ntile * 16 + (lane & 15);
  const int base = ldsMode<MODE>(p0, n);
#pragma unroll
  for (int v = 0; v < 8; ++v)
    cube[base + v * PS<MODE>::v] = d[v];
}

// In-place real contraction: cube <- contract_MODE(cube, W[32][32])
template <int MODE>
__device__ __forceinline__ void contract(float* cube, const float* __restrict__ W,
                                         int tid) {
  const int wave  = tid >> 5, lane = tid & 31;
  const int ntile = wave >> 4;        // 0..1
  const int mbase = (wave & 15) * 4;  // 4 mtiles per wave (64 mtiles total)
  v2f b[8];
  loadB(W, ntile, lane, b);
  v8f acc[4];
#pragma unroll
  for (int t = 0; t < 4; ++t) {
    v8f c = {0.f, 0.f, 0.f, 0.f, 0.f, 0.f, 0.f, 0.f};
    tileAccum<MODE>(cube, mbase + t, lane, b, c);
    acc[t] = c;
  }
  __syncthreads();                    // all reads done before in-place writes
#pragma unroll
  for (int t = 0; t < 4; ++t)
    storeTile<MODE>(cube, mbase + t, ntile, lane, acc[t]);
  __syncthreads();
}

// In-place complex step: (zr,zi) <- (zr*Wr - zi*Wi, zr*Wi + zi*Wr) along MODE
template <int MODE, bool ZI_ZERO>
__device__ __forceinline__ void complexStep(float* zr, float* zi,
                                            const float* __restrict__ Wr,
                                            const float* __restrict__ Wi, int tid) {
  const int wave  = tid >> 5, lane = tid & 31;
  const int ntile = wave >> 4;
  const int mbase = (wave & 15) * 4;
  v2f br[8], bi[8];
  loadB(Wr, ntile, lane, br);
  loadB(Wi, ntile, lane, bi);
  v8f ar[4], ai[4];
#pragma unroll
  for (int t = 0; t < 4; ++t) {
    v8f cr = {0.f, 0.f, 0.f, 0.f, 0.f, 0.f, 0.f, 0.f};
    v8f ci = {0.f, 0.f, 0.f, 0.f, 0.f, 0.f, 0.f, 0.f};
    const int p    = (mbase + t) * 16 + (lane & 15);
    const int qoff = (lane >= 16) ? 2 : 0;
    const int base = ldsMode<MODE>(p, qoff);
#pragma unroll
    for (int k = 0; k < 8; ++k) {
      v2f azr;
      azr.x = zr[base + (4 * k)     * QS<MODE>::v];
      azr.y = zr[base + (4 * k + 1) * QS<MODE>::v];
      cr = wmma4(azr, br[k], cr);
      ci = wmma4(azr, bi[k], ci);
      if (!ZI_ZERO) {
        v2f azi, naz;
        azi.x = zi[base + (4 * k)     * QS<MODE>::v];
        azi.y = zi[base + (4 * k + 1) * QS<MODE>::v];
        naz.x = -azi.x; naz.y = -azi.y;      // f32 WMMA has no A/B-neg modifier
        cr = wmma4(naz, bi[k], cr);
        ci = wmma4(azi, br[k], ci);
      }
    }
    ar[t] = cr; ai[t] = ci;
  }
  __syncthreads();
#pragma unroll
  for (int t = 0; t < 4; ++t) {
    storeTile<MODE>(zr, mbase + t, ntile, lane, ar[t]);
    storeTile<MODE>(zi, mbase + t, ntile, lane, ai[t]);
  }
  __syncthreads();
}

__device__ __forceinline__ void loadCube(float* cube, const float* __restrict__ g,
                                         int tid) {
#pragma unroll
  for (int c = 0; c < 8; ++c) {
    int i = c * 4096 + tid * 4;
    *(float4*)(cube + ldsLin(i)) = *(const float4*)(g + i);
  }
}

__device__ __forceinline__ float sigm(float v) { return 1.f / (1.f + expf(-v)); }

__global__ __launch_bounds__(NT, 1)
void cube_cell_kernel(const float* __restrict__ x, const float* __restrict__ h,
                      const float* __restrict__ acc,
                      const float* __restrict__ ig, const float* __restrict__ rg,
                      const float* __restrict__ cg, const float* __restrict__ dec,
                      const float* __restrict__ cpl, const float* __restrict__ pr,
                      const float* __restrict__ pi, const float* __restrict__ pw,
                      const float* __restrict__ s_in, const float* __restrict__ s_rec,
                      const float* __restrict__ s_carry, const float* __restrict__ s_cpl,
                      float* __restrict__ out) {
  extern __shared__ float lds[];
  float* c0 = lds;               // cube 0 (zr / coupled / pred)
  float* c1 = lds + CUBE_PAD;    // cube 1 (zi)
  __shared__ float redbuf[32];
  __shared__ float inv_sh;

  const int tid = threadIdx.x;
  const size_t base = (size_t)blockIdx.x * CUBE;
  float* outh = out;                       // new_h region (also used as staging)
  float* outa = out + (size_t)256 * CUBE;  // new_acc region

  // ---- phase 1: complex modewise projection, Re part -> outh (staging) ----
  loadCube(c0, h + base, tid);
  __syncthreads();
  complexStep<0, true >(c0, c1, pr,        pi,        tid);
  complexStep<1, false>(c0, c1, pr + 1024, pi + 1024, tid);
  complexStep<2, false>(c0, c1, pr + 2048, pi + 2048, tid);
#pragma unroll
  for (int c = 0; c < 8; ++c) {
    int i = c * 4096 + tid * 4;
    *(float4*)(outh + base + i) = *(const float4*)(c0 + ldsLin(i));
  }
  __syncthreads();

  // ---- phase 2: sequential coupling chain; outh += coupled ----
  loadCube(c0, h + base, tid);
  __syncthreads();
  contract<0>(c0, cpl,        tid);
  contract<1>(c0, cpl + 1024, tid);
  contract<2>(c0, cpl + 2048, tid);
#pragma unroll
  for (int c = 0; c < 8; ++c) {           // same thread wrote these in phase 1
    int i = c * 4096 + tid * 4;
    float4 z = *(const float4*)(outh + base + i);
    float4 q = *(const float4*)(c0 + ldsLin(i));
    z.x += q.x; z.y += q.y; z.z += q.z; z.w += q.w;
    *(float4*)(outh + base + i) = z;
  }
  __syncthreads();

  // ---- phase 3: modewise prediction chain (pred stays in LDS c0) ----
  loadCube(c0, h + base, tid);
  __syncthreads();
  contract<0>(c0, pw,        tid);
  contract<1>(c0, pw + 1024, tid);
  contract<2>(c0, pw + 2048, tid);

  // ---- phase 4: fused elementwise + per-cube RMS + accumulator update ----
  const float si = sigm(*s_in), sr = sigm(*s_rec);
  const float sc = sigm(*s_carry), sp = sigm(*s_cpl);
  float pre[32];
  float sumsq = 0.f;
#pragma unroll
  for (int c = 0; c < 8; ++c) {
    int i = c * 4096 + tid * 4;
    float4 hv  = *(const float4*)(h + base + i);
    float4 xv  = *(const float4*)(x + base + i);
    float4 av  = *(const float4*)(acc + base + i);
    float4 igv = *(const float4*)(ig + i);
    float4 rgv = *(const float4*)(rg + i);
    float4 cgv = *(const float4*)(cg + i);
    float4 dv  = *(const float4*)(dec + i);
    float4 czv = *(const float4*)(outh + base + i);  // coupled + zr
    float4 pv  = *(const float4*)(c0 + ldsLin(i));   // pred
    const float hs[4]  = {hv.x, hv.y, hv.z, hv.w};
    const float xs[4]  = {xv.x, xv.y, xv.z, xv.w};
    const float as[4]  = {av.x, av.y, av.z, av.w};
    const float igs[4] = {igv.x, igv.y, igv.z, igv.w};
    const float rgs[4] = {rgv.x, rgv.y, rgv.z, rgv.w};
    const float cgs[4] = {cgv.x, cgv.y, cgv.z, cgv.w};
    const float ds[4]  = {dv.x, dv.y, dv.z, dv.w};
    const float czs[4] = {czv.x, czv.y, czv.z, czv.w};
    const float ps[4]  = {pv.x, pv.y, pv.z, pv.w};
#pragma unroll
    for (int j = 0; j < 4; ++j) {
      float at = tanhf(as[j]);
      float dg = sigm(ds[j]);
      float m_in = 1.f + at * si, m_rec = 1.f + at * sr;
      float m_ca = 1.f + at * sc, m_cp = 1.f + at * sp;
      float p_ = dg * hs[j] + igs[j] * m_in * xs[j]
               + rgs[j] * m_rec * (czs[j] * m_cp)
               + cgs[j] * m_ca * as[j]
               + ETA * (xs[j] - ps[j]);
      pre[c * 4 + j] = p_;
      sumsq += p_ * p_;
    }
  }
  // wave32 reduce, then block reduce over 32 waves
#pragma unroll
  for (int o = 16; o > 0; o >>= 1) sumsq += __shfl_xor(sumsq, o, 32);
  if ((tid & 31) == 0) redbuf[tid >> 5] = sumsq;
  __syncthreads();
  if (tid < 32) {
    float v = redbuf[tid];
#pragma unroll
    for (int o = 16; o > 0; o >>= 1) v += __shfl_xor(v, o, 32);
    if (tid == 0) inv_sh = rsqrtf(v * (1.0f / (float)CUBE) + 1e-6f);
  }
  __syncthreads();
  const float inv = inv_sh;
#pragma unroll
  for (int c = 0; c < 8; ++c) {
    int i = c * 4096 + tid * 4;
    float4 av = *(const float4*)(acc + base + i);
    const float as[4] = {av.x, av.y, av.z, av.w};
    float4 nh, na;
    float nhs[4], nas[4];
#pragma unroll
    for (int j = 0; j < 4; ++j) {
      nhs[j] = pre[c * 4 + j] * inv;
      nas[j] = MAGD * as[j] + (1.f - MAGD) * fabsf(nhs[j]);
    }
    nh.x = nhs[0]; nh.y = nhs[1]; nh.z = nhs[2]; nh.w = nhs[3];
    na.x = nas[0]; na.y = nas[1]; na.z = nas[2]; na.w = nas[3];
    *(float4*)(outh + base + i) = nh;
    *(float4*)(outa + base + i) = na;
  }
}

extern "C" void kernel_launch(void* const* d_in, const int* in_sizes, int n_in,
                              void* d_out, int out_size, void* d_ws, size_t ws_size,
                              hipStream_t stream) {
  const float* x    = (const float*)d_in[0];
  const float* h    = (const float*)d_in[1];
  const float* acc  = (const float*)d_in[2];
  const float* ig   = (const float*)d_in[3];
  const float* rg   = (const float*)d_in[4];
  const float* cg   = (const float*)d_in[5];
  const float* dec  = (const float*)d_in[6];
  const float* cpl  = (const float*)d_in[7];
  const float* pr   = (const float*)d_in[8];
  const float* pi   = (const float*)d_in[9];
  const float* pw   = (const float*)d_in[10];
  const float* s_in = (const float*)d_in[11];
  const float* s_re = (const float*)d_in[12];
  const float* s_ca = (const float*)d_in[13];
  const float* s_cp = (const float*)d_in[14];
  float* out = (float*)d_out;

  const size_t shmem = (size_t)2 * CUBE_PAD * sizeof(float);  // 295,872 B < 320 KB
  cube_cell_kernel<<<256, NT, shmem, stream>>>(x, h, acc, ig, rg, cg, dec,
                                               cpl, pr, pi, pw,
                                               s_in, s_re, s_ca, s_cp, out);
}